// TimeSliceGNN_89446988906518
// MI455X (gfx1250) — compile-verified
//
#include <hip/hip_runtime.h>
#include <hip/hip_bf16.h>

typedef __attribute__((ext_vector_type(2))) float v2f;
typedef __attribute__((ext_vector_type(8))) float v8f;

__device__ __forceinline__ v8f wmma4(v2f a, v2f b, v8f c) {
    return __builtin_amdgcn_wmma_f32_16x16x4_f32(false, a, false, b, (short)0, c,
                                                 false, false);
}

// ---------------------------------------------------------------------------
// Setup kernels
// ---------------------------------------------------------------------------
__global__ void k_fill(float* __restrict__ p, int n, float v) {
    int i = blockIdx.x * blockDim.x + threadIdx.x;
    if (i < n) p[i] = v;
}

__global__ void k_deg(const int* __restrict__ dst, float* __restrict__ deg, int E) {
    int e = blockIdx.x * blockDim.x + threadIdx.x;
    if (e < E) atomicAdd(&deg[dst[e]], 1.0f);
}

__global__ void k_norm(const float* __restrict__ deg, float* __restrict__ dis,
                       float* __restrict__ dinv, int N) {
    int i = blockIdx.x * blockDim.x + threadIdx.x;
    if (i < N) {
        float d = deg[i];
        dis[i]  = rsqrtf(d);
        dinv[i] = 1.0f / d;
    }
}

__global__ void k_enorm(const int* __restrict__ src, const int* __restrict__ dst,
                        const float* __restrict__ dis, float* __restrict__ enorm, int E) {
    int e = blockIdx.x * blockDim.x + threadIdx.x;
    if (e < E) enorm[e] = dis[src[e]] * dis[dst[e]];
}

__global__ void k_softmax_att(const float* __restrict__ att, float* __restrict__ probs,
                              int T) {
    if (threadIdx.x == 0 && blockIdx.x == 0) {
        float m = -1e30f;
        for (int t = 0; t < T; ++t) m = fmaxf(m, att[t]);
        float s = 0.f;
        for (int t = 0; t < T; ++t) { float p = __expf(att[t] - m); probs[t] = p; s += p; }
        float inv = 1.f / s;
        for (int t = 0; t < T; ++t) probs[t] *= inv;
    }
}

// ---------------------------------------------------------------------------
// Shared GEMM epilogue: hw = acc ; gout = dinv*acc + bias  (self-loop fused)
// C/D layout: VGPR j -> M = row0 + j + 8*(lane>=16), N = lane&15.
// ---------------------------------------------------------------------------
__device__ __forceinline__ void gemm_epilogue(v8f acc, float* __restrict__ hw,
                                              float* __restrict__ gout,
                                              const float* __restrict__ dinv,
                                              float bl, int row0, int half, int l) {
    const int mbase = row0 + 8 * half;
#pragma unroll
    for (int j = 0; j < 8; ++j) {
        const int m  = mbase + j;
        const float v = acc[j];
        hw[(size_t)m * 16 + l]   = v;
        gout[(size_t)m * 16 + l] = dinv[m] * v + bl;
    }
}

// ---------------------------------------------------------------------------
// K=10 GEMM (input features), zero-padded to 12. Unconditional vector loads;
// the only pad (k=10,11 on lanes 16-31) is a register select after an
// in-bounds load (a==0 makes the stale b rows harmless: 0*finite == 0).
// ---------------------------------------------------------------------------
__global__ void k_gemm_fin(const float* __restrict__ X, const float* __restrict__ W,
                           float* __restrict__ hw, float* __restrict__ gout,
                           const float* __restrict__ dinv, const float* __restrict__ bias,
                           int N) {
    const int lane = threadIdx.x & 31;
    const int wave = threadIdx.x >> 5;
    const int row0 = (blockIdx.x * (blockDim.x >> 5) + wave) * 16;
    if (row0 >= N) return;                 // wave-uniform
    const int half = lane >> 4;
    const int l    = lane & 15;
    const float* xrow = X + (size_t)(row0 + l) * 10;
    v8f acc = {};
#pragma unroll
    for (int k0 = 0; k0 < 8; k0 += 4) {    // k0 = 0, 4 : fully in-bounds
        const int ka = k0 + 2 * half;
        v2f a = *(const v2f*)(xrow + ka);
        v2f b;
        b.x = W[(size_t)ka * 16 + l];
        b.y = W[(size_t)ka * 16 + 16 + l];
        acc = wmma4(a, b, acc);
    }
    {                                      // k0 = 8 : K=10,11 padded to zero
        v2f a = *(const v2f*)(xrow + 8);   // always in-bounds load
        a.x = half ? 0.f : a.x;
        a.y = half ? 0.f : a.y;
        v2f b;                             // rows 8,9 for everyone (a==0 on half=1)
        b.x = W[8 * 16 + l];
        b.y = W[9 * 16 + l];
        acc = wmma4(a, b, acc);
    }
    gemm_epilogue(acc, hw, gout, dinv, bias[l], row0, half, l);
}

// ---------------------------------------------------------------------------
// K=16 GEMM, optional relu on A load (compile-time), fused self-loop epilogue.
// ---------------------------------------------------------------------------
template <bool RELU>
__global__ void k_gemm16(const float* __restrict__ X, const float* __restrict__ W,
                         float* __restrict__ hw, float* __restrict__ gout,
                         const float* __restrict__ dinv, const float* __restrict__ bias,
                         int N) {
    const int lane = threadIdx.x & 31;
    const int wave = threadIdx.x >> 5;
    const int row0 = (blockIdx.x * (blockDim.x >> 5) + wave) * 16;
    if (row0 >= N) return;
    const int half = lane >> 4;
    const int l    = lane & 15;
    const float* xrow = X + (size_t)(row0 + l) * 16;
    v8f acc = {};
#pragma unroll
    for (int k0 = 0; k0 < 16; k0 += 4) {
        const int ka = k0 + 2 * half;
        v2f a = *(const v2f*)(xrow + ka);
        if (RELU) { a.x = fmaxf(a.x, 0.f); a.y = fmaxf(a.y, 0.f); }
        v2f b;
        b.x = W[(size_t)ka * 16 + l];
        b.y = W[(size_t)ka * 16 + 16 + l];
        acc = wmma4(a, b, acc);
    }
    gemm_epilogue(acc, hw, gout, dinv, bias[l], row0, half, l);
}

// ---------------------------------------------------------------------------
// Gate GEMM: out = act( [G | H(*Rg)] [N x 32] @ L[32 x 16] + bias )
// concat fused structurally (first 4 k-steps read G, last 4 read H);
// HAS_RG / ACT are compile-time. ACT: 1=sigmoid, 2=tanh.
// ---------------------------------------------------------------------------
template <bool HAS_RG, int ACT>
__global__ void k_gate(const float* __restrict__ G, const float* __restrict__ H,
                       const float* __restrict__ Rg, const float* __restrict__ L,
                       const float* __restrict__ bias, float* __restrict__ out, int N) {
    const int lane = threadIdx.x & 31;
    const int wave = threadIdx.x >> 5;
    const int row0 = (blockIdx.x * (blockDim.x >> 5) + wave) * 16;
    if (row0 >= N) return;
    const int half = lane >> 4;
    const int l    = lane & 15;
    const size_t rb = (size_t)(row0 + l) * 16;
    v8f acc = {};
#pragma unroll
    for (int k0 = 0; k0 < 16; k0 += 4) {   // A cols 0..15  -> G
        const int ka = k0 + 2 * half;
        v2f a = *(const v2f*)(G + rb + ka);
        v2f b;
        b.x = L[(size_t)ka * 16 + l];
        b.y = L[(size_t)ka * 16 + 16 + l];
        acc = wmma4(a, b, acc);
    }
#pragma unroll
    for (int k0 = 16; k0 < 32; k0 += 4) {  // A cols 16..31 -> H (* Rg)
        const int ka = k0 + 2 * half;
        const int c  = ka - 16;
        v2f a = *(const v2f*)(H + rb + c);
        if (HAS_RG) {
            v2f m = *(const v2f*)(Rg + rb + c);
            a.x *= m.x; a.y *= m.y;
        }
        v2f b;
        b.x = L[(size_t)ka * 16 + l];
        b.y = L[(size_t)ka * 16 + 16 + l];
        acc = wmma4(a, b, acc);
    }
    const float bl = bias[l];
    const int mbase = row0 + 8 * half;
#pragma unroll
    for (int j = 0; j < 8; ++j) {
        float v = acc[j] + bl;
        v = (ACT == 1) ? (1.f / (1.f + __expf(-v))) : tanhf(v);
        out[(size_t)(mbase + j) * 16 + l] = v;
    }
}

// ---------------------------------------------------------------------------
// Edge scatter: out[dst] += enorm * hw[src]   (16 lanes per edge)
// ---------------------------------------------------------------------------
__global__ void k_scatter(const float* __restrict__ hw, const float* __restrict__ enorm,
                          const int* __restrict__ src, const int* __restrict__ dst,
                          float* __restrict__ out, int total /* = E*16 */) {
    int idx = blockIdx.x * blockDim.x + threadIdx.x;
    if (idx >= total) return;
    int e = idx >> 4, f = idx & 15;
    int s = src[e], d = dst[e];
    atomicAdd(&out[(size_t)d * 16 + f], enorm[e] * hw[(size_t)s * 16 + f]);
}

// ---------------------------------------------------------------------------
// float4 elementwise kernels
// ---------------------------------------------------------------------------
__global__ void k_relu4(const float4* __restrict__ in, float4* __restrict__ out, int n4) {
    int i = blockIdx.x * blockDim.x + threadIdx.x;
    if (i < n4) {
        float4 v = in[i];
        v.x = fmaxf(v.x, 0.f); v.y = fmaxf(v.y, 0.f);
        v.z = fmaxf(v.z, 0.f); v.w = fmaxf(v.w, 0.f);
        out[i] = v;
    }
}

// H = Z*H + (1-Z)*Ht ;  Hacc += probs[t]*H
__global__ void k_gru_update4(const float4* __restrict__ Zg, const float4* __restrict__ Ht,
                              float4* __restrict__ H, float4* __restrict__ Hacc,
                              const float* __restrict__ probs, int t, int n4) {
    int i = blockIdx.x * blockDim.x + threadIdx.x;
    if (i >= n4) return;
    const float p = probs[t];
    float4 z = Zg[i], ht = Ht[i], h = H[i], a = Hacc[i];
    h.x = z.x * h.x + (1.f - z.x) * ht.x;
    h.y = z.y * h.y + (1.f - z.y) * ht.y;
    h.z = z.z * h.z + (1.f - z.z) * ht.z;
    h.w = z.w * h.w + (1.f - z.w) * ht.w;
    H[i] = h;
    a.x += p * h.x; a.y += p * h.y; a.z += p * h.z; a.w += p * h.w;
    Hacc[i] = a;
}

// logits = Hacc @ fcW + fcb ; log_softmax over C=2
__global__ void k_head(const float* __restrict__ Hacc, const float* __restrict__ fcW,
                       const float* __restrict__ fcb, float* __restrict__ out, int N) {
    int i = blockIdx.x * blockDim.x + threadIdx.x;
    if (i >= N) return;
    const float* h = Hacc + (size_t)i * 16;
    float l0 = fcb[0], l1 = fcb[1];
#pragma unroll
    for (int f = 0; f < 16; ++f) {
        l0 += h[f] * fcW[f * 2 + 0];
        l1 += h[f] * fcW[f * 2 + 1];
    }
    float m   = fmaxf(l0, l1);
    float lse = m + logf(__expf(l0 - m) + __expf(l1 - m));
    out[(size_t)i * 2 + 0] = l0 - lse;
    out[(size_t)i * 2 + 1] = l1 - lse;
}

// ---------------------------------------------------------------------------
static inline int cdiv(long long a, long long b) { return (int)((a + b - 1) / b); }

extern "C" void kernel_launch(void* const* d_in, const int* in_sizes, int n_in,
                              void* d_out, int out_size, void* d_ws, size_t ws_size,
                              hipStream_t stream) {
    const float* x   = (const float*)d_in[0];
    const int*   src = (const int*)d_in[1];
    const int*   dst = (const int*)d_in[2];
    const float* W1  = (const float*)d_in[3];
    const float* b1  = (const float*)d_in[4];
    const float* W2  = (const float*)d_in[5];
    const float* b2  = (const float*)d_in[6];
    const float* Wz  = (const float*)d_in[7];
    const float* bz  = (const float*)d_in[8];
    const float* Wr  = (const float*)d_in[9];
    const float* br  = (const float*)d_in[10];
    const float* Wh  = (const float*)d_in[11];
    const float* bh  = (const float*)d_in[12];
    const float* Lz  = (const float*)d_in[13];
    const float* lz  = (const float*)d_in[14];
    const float* Lr  = (const float*)d_in[15];
    const float* lr  = (const float*)d_in[16];
    const float* Lh  = (const float*)d_in[17];
    const float* lh  = (const float*)d_in[18];
    const float* att = (const float*)d_in[19];
    const float* fcW = (const float*)d_in[20];
    const float* fcb = (const float*)d_in[21];

    const int T   = in_sizes[19];
    const int HID = in_sizes[4];            // 16
    const int FIN = in_sizes[3] / HID;      // 10
    const int E   = in_sizes[1];
    const int N   = in_sizes[0] / (T * FIN);
    const int NH  = N * HID;                // N*16 elements
    const int ET  = E * 16;

    // workspace carve-up (floats); all offsets stay 16B-aligned
    float* ws     = (float*)d_ws;
    float* deg    = ws;  ws += N;
    float* dis    = ws;  ws += N;
    float* dinv   = ws;  ws += N;
    float* enorm  = ws;  ws += E;
    float* probs  = ws;  ws += 8;
    float* hw     = ws;  ws += (size_t)NH;
    float* gcnbuf = ws;  ws += (size_t)NH;
    float* feats  = ws;  ws += (size_t)T * NH;
    float* H      = ws;  ws += (size_t)NH;
    float* Zg     = ws;  ws += (size_t)NH;
    float* Rg     = ws;  ws += (size_t)NH;
    float* Ht     = ws;  ws += (size_t)NH;
    float* Hacc   = ws;  ws += (size_t)NH;

    const int B     = 256;
    const int gGemm = cdiv(N, 128);         // 8 waves/block * 16 rows/wave
    const int gNH4  = cdiv(NH / 4, B);
    const int gET   = cdiv(ET, B);

    // ---- normalization setup ----
    k_fill<<<cdiv(N, B), B, 0, stream>>>(deg, N, 1.0f);   // self-loop
    k_deg<<<cdiv(E, B), B, 0, stream>>>(dst, deg, E);
    k_norm<<<cdiv(N, B), B, 0, stream>>>(deg, dis, dinv, N);
    k_enorm<<<cdiv(E, B), B, 0, stream>>>(src, dst, dis, enorm, E);
    k_softmax_att<<<1, 32, 0, stream>>>(att, probs, T);
    k_fill<<<cdiv(NH, B), B, 0, stream>>>(H, NH, 0.0f);
    k_fill<<<cdiv(NH, B), B, 0, stream>>>(Hacc, NH, 0.0f);

    // GCN with K=16 input: gcnbuf = deg_inv*(A@W) + b, then scatter edges
    auto gcn16 = [&](const float* A, const float* W, const float* b) {
        k_gemm16<false><<<gGemm, B, 0, stream>>>(A, W, hw, gcnbuf, dinv, b, N);
        k_scatter<<<gET, B, 0, stream>>>(hw, enorm, src, dst, gcnbuf, ET);
    };

    // ---- per-slice 2-layer GCN feature extraction ----
    for (int t = 0; t < T; ++t) {
        const float* xt = x + (size_t)t * N * FIN;
        // layer 1 (K = FIN)
        k_gemm_fin<<<gGemm, B, 0, stream>>>(xt, W1, hw, gcnbuf, dinv, b1, N);
        k_scatter<<<gET, B, 0, stream>>>(hw, enorm, src, dst, gcnbuf, ET);
        // layer 2: relu fused into A load; epilogue overwrites gcnbuf in place
        k_gemm16<true><<<gGemm, B, 0, stream>>>(gcnbuf, W2, hw, gcnbuf, dinv, b2, N);
        k_scatter<<<gET, B, 0, stream>>>(hw, enorm, src, dst, gcnbuf, ET);
        k_relu4<<<gNH4, B, 0, stream>>>((const float4*)gcnbuf,
                                        (float4*)(feats + (size_t)t * NH), NH / 4);
    }

    // ---- TGCN GRU over time ----
    for (int t = 0; t < T; ++t) {
        const float* ft = feats + (size_t)t * NH;
        gcn16(ft, Wz, bz);
        k_gate<false, 1><<<gGemm, B, 0, stream>>>(gcnbuf, H, nullptr, Lz, lz, Zg, N);
        gcn16(ft, Wr, br);
        k_gate<false, 1><<<gGemm, B, 0, stream>>>(gcnbuf, H, nullptr, Lr, lr, Rg, N);
        gcn16(ft, Wh, bh);
        k_gate<true, 2><<<gGemm, B, 0, stream>>>(gcnbuf, H, Rg, Lh, lh, Ht, N);
        k_gru_update4<<<gNH4, B, 0, stream>>>((const float4*)Zg, (const float4*)Ht,
                                              (float4*)H, (float4*)Hacc, probs, t,
                                              NH / 4);
    }

    // ---- attention-accumulated head ----
    k_head<<<cdiv(N, B), B, 0, stream>>>(Hacc, fcW, fcb, (float*)d_out, N);
}